// VoxelNet_19318762898067
// MI455X (gfx1250) — compile-verified
//
#include <hip/hip_runtime.h>
#include <hip/hip_bf16.h>
#include <stdint.h>

// ---------------- problem constants (match reference) ----------------
#define GXD 1024u
#define GYD 1024u
#define GZD 40u
#define NBATCH 4u
#define TOTAL_CELLS (NBATCH * GXD * GYD * GZD)   // 167,772,160 cells
#define NWORDS (TOTAL_CELLS / 32u)               // 5,242,880 bitmap words (21 MB, L2-resident)
#define WORDS_PER_BLOCK 1024u
#define SCAN_BLOCKS (NWORDS / WORDS_PER_BLOCK)   // 5120 (exact)

#define TILE_PTS 1024
#define TILE_FLOATS (TILE_PTS * 5)               // 20 KB LDS tile

// types matching gfx1250 builtin prototypes (verified via hipcc diagnostics)
typedef int v4i_ __attribute__((vector_size(16)));
typedef unsigned int u32x4_ __attribute__((vector_size(16)));
typedef int i32x4_ __attribute__((vector_size(16)));
typedef int i32x8_ __attribute__((vector_size(32)));
#define AS1P(T) T __attribute__((address_space(1))) *
#define AS3P(T) T __attribute__((address_space(3))) *

#define HAVE_TDM (defined(__gfx1250__) && \
                  __has_builtin(__builtin_amdgcn_tensor_load_to_lds) && \
                  __has_builtin(__builtin_amdgcn_s_wait_tensorcnt))
#define HAVE_ASYNC (defined(__gfx1250__) && \
                    __has_builtin(__builtin_amdgcn_global_load_async_to_lds_b128) && \
                    __has_builtin(__builtin_amdgcn_global_load_async_to_lds_b32))

// lexicographic key over (batch, cx, cy, cz); ascending key == jnp.unique row order
__device__ __forceinline__ uint32_t voxel_key(float b, float x, float y, float z) {
    // replicate reference f32 math exactly: (p - lo) / vs, truncate toward zero
    float cxf = (x - (-51.2f)) / 0.1f;
    float cyf = (y - (-51.2f)) / 0.1f;
    float czf = (z - (-5.0f)) / 0.2f;
    int cx = (int)cxf, cy = (int)cyf, cz = (int)czf;
    cx = min(max(cx, 0), (int)GXD - 1);
    cy = min(max(cy, 0), (int)GYD - 1);
    cz = min(max(cz, 0), (int)GZD - 1);
    int bi = min(max((int)b, 0), (int)NBATCH - 1);
    return (((uint32_t)bi * GXD + (uint32_t)cx) * GYD + (uint32_t)cy) * GZD + (uint32_t)cz;
}

// ---- K1: TDM-stage point tile to LDS, compute key, mark bitmap, emit features ----
__global__ __launch_bounds__(256)
void key_mark_kernel(const float* __restrict__ pts, uint32_t* __restrict__ bitmap,
                     uint32_t* __restrict__ keys, float* __restrict__ feat_out, int n)
{
    __shared__ __align__(16) float tile[TILE_FLOATS];
    int tile0 = blockIdx.x * TILE_PTS;
    int npts = n - tile0;
    if (npts <= 0) return;
    if (npts > TILE_PTS) npts = TILE_PTS;
    int nflt = npts * 5;
    const float* gsrc = pts + (size_t)tile0 * 5;

#if HAVE_TDM
    // One Tensor Data Mover descriptor DMAs the whole 20KB tile to LDS
    // (CDNA5 ISA): 1D tile of nflt 4-byte elements.
    if (threadIdx.x == 0u) {
        uint32_t ldsOff = (uint32_t)(uintptr_t)(AS3P(float))tile;
        uint64_t ga = (uint64_t)(uintptr_t)gsrc;
        uint32_t nf = (uint32_t)nflt;
        // Group 0: count=1 | lds_addr | global_addr[56:0] | type=2
        u32x4_ g0 = { 1u,
                      ldsOff,
                      (uint32_t)ga,
                      (uint32_t)((ga >> 32) & 0x01FFFFFFu) | (2u << 30) };
        // Group 1: data_size=4B; tensor_dim0=nf; tensor_dim1=1; tile_dim0=nf;
        //          tile_dim1=1; tensor_dim0_stride=nf; all padding/multicast off.
        i32x8_ g1 = { (int)(2u << 16),                       // data_size=2 (4 bytes)
                      (int)((nf & 0xFFFFu) << 16),           // tensor_dim0[15:0]
                      (int)((nf >> 16) | (1u << 16)),        // tensor_dim0[31:16] | tensor_dim1 lo
                      (int)(nf << 16),                       // tensor_dim1 hi(0) | tile_dim0=nf
                      1,                                     // tile_dim1=1, tile_dim2=0
                      (int)nf,                               // tensor_dim0_stride[31:0]
                      0, 0 };                                // stride hi, tensor_dim1_stride
        i32x4_ gz = { 0, 0, 0, 0 };
#if __has_include(<hip/amd_detail/amd_gfx1250_TDM.h>)
        i32x8_ gz8 = { 0, 0, 0, 0, 0, 0, 0, 0 };
        __builtin_amdgcn_tensor_load_to_lds(g0, g1, gz, gz, gz8, 0);   // 6-arg toolchain
#else
        __builtin_amdgcn_tensor_load_to_lds(g0, g1, gz, gz, 0);        // 5-arg (ROCm 7.2)
#endif
        __builtin_amdgcn_s_wait_tensorcnt(0);
    }
    __syncthreads();
#elif HAVE_ASYNC
    // per-lane async DMA fallback: coalesced 16B/lane global -> LDS (ASYNCcnt)
    int nchunk = nflt >> 2;
    for (int c = (int)threadIdx.x; c < nchunk; c += (int)blockDim.x) {
        __builtin_amdgcn_global_load_async_to_lds_b128(
            (AS1P(v4i_))(gsrc + 4 * c),
            (AS3P(v4i_))(tile + 4 * c), 0, 0);
    }
    for (int w = (nchunk << 2) + (int)threadIdx.x; w < nflt; w += (int)blockDim.x) {
        __builtin_amdgcn_global_load_async_to_lds_b32(
            (AS1P(int))(gsrc + w),
            (AS3P(int))(tile + w), 0, 0);
    }
#if __has_builtin(__builtin_amdgcn_s_wait_asynccnt)
    __builtin_amdgcn_s_wait_asynccnt(0);
#else
    asm volatile("s_wait_asynccnt 0" ::: "memory");
#endif
    __syncthreads();
#else
    for (int w = (int)threadIdx.x; w < nflt; w += (int)blockDim.x) tile[w] = gsrc[w];
    __syncthreads();
#endif

    // speculative prefetch of the next tile (global_prefetch_b8)
    if ((tile0 + TILE_PTS) < n && threadIdx.x < 64) {
        __builtin_prefetch(pts + (size_t)(tile0 + TILE_PTS) * 5 + threadIdx.x * 32, 0, 3);
    }

    for (int p = (int)threadIdx.x; p < npts; p += (int)blockDim.x) {
        float b = tile[p * 5 + 0];   // stride-5 over 64 banks: conflict-free
        float x = tile[p * 5 + 1];
        float y = tile[p * 5 + 2];
        float z = tile[p * 5 + 3];
        float f = tile[p * 5 + 4];
        uint32_t k = voxel_key(b, x, y, z);
        int gi = tile0 + p;
        keys[gi] = k;
        atomicOr(&bitmap[k >> 5], 1u << (k & 31u));   // 21MB bitmap: stays in 192MB L2
        float4 fo = make_float4(x, y, z, f);
        *reinterpret_cast<float4*>(feat_out + (size_t)gi * 4) = fo;
    }
}

// ---- K2: per-block popcount exclusive scan over 1024 words/block ----
__global__ __launch_bounds__(256)
void scan1_kernel(const uint32_t* __restrict__ bitmap, uint32_t* __restrict__ wordPrefix,
                  uint32_t* __restrict__ blockSums)
{
    __shared__ uint32_t sh[256];
    uint32_t t = threadIdx.x;
    uint32_t base = blockIdx.x * WORDS_PER_BLOCK + t * 4u;  // NWORDS % 1024 == 0, no guards needed
    uint4 w = *reinterpret_cast<const uint4*>(bitmap + base);
    uint32_t c0 = __popc(w.x), c1 = __popc(w.y), c2 = __popc(w.z), c3 = __popc(w.w);
    uint32_t sum = c0 + c1 + c2 + c3;
    sh[t] = sum;
    __syncthreads();
    for (int off = 1; off < 256; off <<= 1) {
        uint32_t v = (t >= (uint32_t)off) ? sh[t - off] : 0u;
        __syncthreads();
        sh[t] += v;
        __syncthreads();
    }
    uint32_t incl = sh[t];
    uint32_t excl = incl - sum;
    if (t == 255u) blockSums[blockIdx.x] = incl;
    uint4 o;
    o.x = excl;
    o.y = excl + c0;
    o.z = excl + c0 + c1;
    o.w = excl + c0 + c1 + c2;
    *reinterpret_cast<uint4*>(wordPrefix + base) = o;
}

// ---- K3: single-block exclusive scan over SCAN_BLOCKS block sums ----
__global__ __launch_bounds__(256)
void scan2_kernel(uint32_t* __restrict__ blockSums, int nb)
{
    __shared__ uint32_t sh[256];
    __shared__ uint32_t carry;
    uint32_t t = threadIdx.x;
    if (t == 0u) carry = 0u;
    __syncthreads();
    for (int base = 0; base < nb; base += 256) {
        int i = base + (int)t;
        uint32_t v = (i < nb) ? blockSums[i] : 0u;
        sh[t] = v;
        __syncthreads();
        for (int off = 1; off < 256; off <<= 1) {
            uint32_t u = (t >= (uint32_t)off) ? sh[t - off] : 0u;
            __syncthreads();
            sh[t] += u;
            __syncthreads();
        }
        uint32_t incl = sh[t];
        uint32_t myCarry = carry;
        if (i < nb) blockSums[i] = incl - v + myCarry;
        __syncthreads();
        if (t == 255u) carry = myCarry + incl;
        __syncthreads();
    }
}

// ---- K4: add scanned block sums into word prefixes ----
__global__ __launch_bounds__(256)
void fixup_kernel(uint32_t* __restrict__ wordPrefix, const uint32_t* __restrict__ blockSums)
{
    for (uint32_t w = blockIdx.x * blockDim.x + threadIdx.x; w < NWORDS;
         w += gridDim.x * blockDim.x)
        wordPrefix[w] += blockSums[w >> 10];
}

// ---- K5: inverse index = rank of key among set bits ----
__global__ __launch_bounds__(256)
void inv_kernel(const uint32_t* __restrict__ keys, const uint32_t* __restrict__ bitmap,
                const uint32_t* __restrict__ wordPrefix, float* __restrict__ inv_out, int n)
{
    int i = (int)(blockIdx.x * blockDim.x + threadIdx.x);
    if (i >= n) return;
    uint32_t k = keys[i];
    uint32_t w = k >> 5, b = k & 31u;
    uint32_t rank = wordPrefix[w] + (uint32_t)__popc(bitmap[w] & ((1u << b) - 1u));
    inv_out[i] = (float)rank;   // ranks < 2^24: exact in f32
}

// ---- K6: emit sorted unique rows as (batch, cz, cy, cx) ----
__global__ __launch_bounds__(256)
void unq_kernel(const uint32_t* __restrict__ bitmap, const uint32_t* __restrict__ wordPrefix,
                float* __restrict__ unq_out, uint32_t U)
{
    for (uint32_t w = blockIdx.x * blockDim.x + threadIdx.x; w < NWORDS;
         w += gridDim.x * blockDim.x) {
        uint32_t m = bitmap[w];
        if (!m) continue;
        uint32_t base = wordPrefix[w];
        while (m) {
            uint32_t b = (uint32_t)__ffs(m) - 1u;
            m &= m - 1u;
            uint32_t key = (w << 5) | b;
            uint32_t cz = key % GZD;
            uint32_t t1 = key / GZD;
            uint32_t cy = t1 % GYD;
            uint32_t t2 = t1 / GYD;
            uint32_t cx = t2 % GXD;
            uint32_t bi = t2 / GXD;
            uint32_t idx = base++;
            if (idx < U) {
                float4 r = make_float4((float)bi, (float)cz, (float)cy, (float)cx);
                *reinterpret_cast<float4*>(unq_out + (size_t)idx * 4) = r;
            }
        }
    }
}

// ---- K7: grid_size output [GZ, GY, GX] ----
__global__ void tail_kernel(float* __restrict__ outGrid)
{
    if (threadIdx.x == 0u) {
        outGrid[0] = (float)GZD;
        outGrid[1] = (float)GYD;
        outGrid[2] = (float)GXD;
    }
}

extern "C" void kernel_launch(void* const* d_in, const int* in_sizes, int n_in,
                              void* d_out, int out_size, void* d_ws, size_t ws_size,
                              hipStream_t stream)
{
    (void)n_in; (void)ws_size;
    const float* pts = (const float*)d_in[0];
    int n = in_sizes[0] / 5;
    float* outF = (float*)d_out;

    // out layout: features (n*4) | unq (U*4) | unq_inv (n) | grid (3)
    long remL = (long)out_size - (long)4 * n - (long)n - 3L;
    uint32_t U = (remL > 0) ? (uint32_t)(remL / 4) : 0u;

    uint32_t* bitmap     = (uint32_t*)d_ws;
    uint32_t* wordPrefix = bitmap + NWORDS;
    uint32_t* blockSums  = wordPrefix + NWORDS;
    uint32_t* keys       = blockSums + (uint32_t)((SCAN_BLOCKS + 255u) & ~255u);

    float* featOut = outF;
    float* unqOut  = outF + (size_t)4 * n;
    float* invOut  = unqOut + (size_t)4 * U;
    float* gridOut = outF + (size_t)out_size - 3;

    (void)hipMemsetAsync(bitmap, 0, (size_t)NWORDS * sizeof(uint32_t), stream);

    int nTiles = (n + TILE_PTS - 1) / TILE_PTS;
    hipLaunchKernelGGL(key_mark_kernel, dim3(nTiles), dim3(256), 0, stream,
                       pts, bitmap, keys, featOut, n);
    hipLaunchKernelGGL(scan1_kernel, dim3(SCAN_BLOCKS), dim3(256), 0, stream,
                       bitmap, wordPrefix, blockSums);
    hipLaunchKernelGGL(scan2_kernel, dim3(1), dim3(256), 0, stream,
                       blockSums, (int)SCAN_BLOCKS);
    hipLaunchKernelGGL(fixup_kernel, dim3(2048), dim3(256), 0, stream,
                       wordPrefix, blockSums);
    hipLaunchKernelGGL(inv_kernel, dim3((n + 255) / 256), dim3(256), 0, stream,
                       keys, bitmap, wordPrefix, invOut, n);
    hipLaunchKernelGGL(unq_kernel, dim3(2048), dim3(256), 0, stream,
                       bitmap, wordPrefix, unqOut, U);
    hipLaunchKernelGGL(tail_kernel, dim3(1), dim3(32), 0, stream, gridOut);
}